// RF_29326036697292
// MI455X (gfx1250) — compile-verified
//
#include <hip/hip_runtime.h>
#include <math.h>

// Resonate-and-fire neuron scan: x[T,B,H] -> (spike[T,B,H], mem[T,B,H]).
// Bandwidth-bound (≈192 MiB traffic, ≈0.2 GFLOP → ≈8.6 µs floor @ 23.3 TB/s).
// CDNA5 path used: GLOBAL_LOAD_ASYNC_TO_LDS_B32 + S_WAIT_ASYNCCNT software
// pipeline (depth 4, 8-slot LDS ring, per-lane slots so no barriers needed).

namespace {
constexpr int   kT       = 128;
constexpr int   kB       = 64;
constexpr int   kH       = 2048;
constexpr int   kN       = kB * kH;        // 131072 independent neurons
constexpr int   kThreads = 256;            // 8 wave32s per block
constexpr int   kBlocks  = kN / kThreads;  // 512 blocks
constexpr float kDT      = 0.01f;
constexpr float kVthr    = 1.0f;
constexpr int   kDepth   = 4;              // async prefetch depth
constexpr int   kRing    = 8;              // LDS ring slots (2x depth)
}  // namespace

#if defined(__HIP_DEVICE_COMPILE__) && defined(__gfx1250__) &&                 \
    __has_builtin(__builtin_amdgcn_global_load_async_to_lds_b32) &&            \
    __has_builtin(__builtin_amdgcn_s_wait_asynccnt)
#define RF_ASYNC_LDS 1
#else
#define RF_ASYNC_LDS 0
#endif

#if RF_ASYNC_LDS
// Builtin signature (from hipcc diagnostic): pointee type is `int`, with
// global (as1) source and LDS (as3) destination; trailing imm offset + cpol.
typedef __attribute__((address_space(1))) int gint_t;
typedef __attribute__((address_space(3))) int lint_t;
__device__ __forceinline__ void rf_async_b32(const float* g, float* l) {
  // Per-lane global address -> per-lane LDS address; tracked by ASYNCcnt.
  __builtin_amdgcn_global_load_async_to_lds_b32((gint_t*)g, (lint_t*)l,
                                                /*offset=*/0, /*cpol=*/0);
}
#endif

// One recurrence step + stores. VTHR == 1 so mem = max(nu, 0), reset = u - z.
#define RF_STEP(XT, TT)                                                        \
  {                                                                            \
    const float ur = u - z * kVthr;                                            \
    const float t1 = fmaf(ur, bco, -(v * om));  /* ur*b - v*omega   */         \
    const float t2 = fmaf(ur, om, v * bco);     /* ur*omega + v*b   */         \
    const float nu = fmaf((XT), kDT, fmaf(t1, kDT, ur));                       \
    const float nv = fmaf(t2, kDT, v);                                         \
    z = (nu > kVthr) ? 1.0f : 0.0f;                                            \
    sp[(size_t)(TT)*kN] = z;                                                   \
    mp[(size_t)(TT)*kN] = fmaxf(nu, 0.0f);                                     \
    u = nu;                                                                    \
    v = nv;                                                                    \
  }

__global__ __launch_bounds__(kThreads) void rf_scan_kernel(
    const float* __restrict__ x, const float* __restrict__ omega,
    const float* __restrict__ b_offset, float* __restrict__ out) {
  const int tid = threadIdx.x;
  const int idx = blockIdx.x * kThreads + tid;  // (b,h) flat index
  const int h   = idx & (kH - 1);               // contiguous per block

  // Per-channel constants (computed once; loop carries only u,v,z in VGPRs).
  const float om   = fabsf(omega[h]);
  const float dtom = kDT * om;
  const float pom  = (-1.0f + sqrtf(1.0f - dtom * dtom)) / kDT;
  const float bco  = pom - fabsf(b_offset[h]);

  const float* xp = x + idx;
  float* sp = out + idx;                     // spikes: [0, T*N)
  float* mp = out + (size_t)kT * kN + idx;   // mem:    [T*N, 2*T*N)

  float u = 0.0f, v = 0.0f, z = 0.0f;

#if RF_ASYNC_LDS
  __shared__ float xbuf[kRing][kThreads];  // 8 KiB ring, per-lane slots

#pragma unroll
  for (int k = 0; k < kDepth; ++k)
    rf_async_b32(xp + (size_t)k * kN, &xbuf[k][tid]);

  // Steady state: slot t is complete once <= kDepth-1 async ops remain.
  for (int t = 0; t < kT - kDepth; ++t) {
    __builtin_amdgcn_s_wait_asynccnt(kDepth - 1);
    const float xt = xbuf[t & (kRing - 1)][tid];
    rf_async_b32(xp + (size_t)(t + kDepth) * kN,
                 &xbuf[(t + kDepth) & (kRing - 1)][tid]);
    RF_STEP(xt, t)
  }

  // Drain: all remaining prefetches must have landed (imm must be constant).
  __builtin_amdgcn_s_wait_asynccnt(0);
#pragma unroll
  for (int t = kT - kDepth; t < kT; ++t) {
    const float xt = xbuf[t & (kRing - 1)][tid];
    RF_STEP(xt, t)
  }
#else
  // Fallback (no async builtins): plain coalesced global loads.
  for (int t = 0; t < kT; ++t) {
    const float xt = xp[(size_t)t * kN];
    RF_STEP(xt, t)
  }
#endif
}

extern "C" void kernel_launch(void* const* d_in, const int* in_sizes, int n_in,
                              void* d_out, int out_size, void* d_ws,
                              size_t ws_size, hipStream_t stream) {
  const float* x        = (const float*)d_in[0];  // [T,B,H] f32
  const float* omega    = (const float*)d_in[1];  // [H]     f32
  const float* b_offset = (const float*)d_in[2];  // [H]     f32
  float* out            = (float*)d_out;          // [2,T,B,H] f32

  rf_scan_kernel<<<kBlocks, kThreads, 0, stream>>>(x, omega, b_offset, out);

  (void)in_sizes; (void)n_in; (void)out_size; (void)d_ws; (void)ws_size;
}